// EpisodicMemory_20907900797332
// MI455X (gfx1250) — compile-verified
//
#include <hip/hip_runtime.h>

#define KDIM 512
#define CDIM 896
#define HDIM 448
#define G4   1792   // 4*HDIM
#define EDIM 4
#define BDIM 8
#define ALPHA_PINV 5e-4f
#define OBS_STD 0.1f

typedef __attribute__((ext_vector_type(2))) float v2f;
typedef __attribute__((ext_vector_type(8))) float v8f;

__device__ __forceinline__ float sigm(float x) { return 1.0f / (1.0f + expf(-x)); }
__device__ __forceinline__ float clampf(float v, float lo, float hi) {
    return fminf(fmaxf(v, lo), hi);
}

// ---------------------------------------------------------------------------
// Batched GEMM with both operands K-major:
//   D[b] = alpha * A[b] @ B[b] + beta * Cin[b]
// A  : (batch, M, Kd)  row-major (K contiguous)
// Bt : (batch, N, Kd)  = B^T, row-major (K contiguous)  -> all frag loads are b64
// D/Cin: (batch, M, N) row-major
// One wave32 per 64x64 output tile: 4x4 grid of V_WMMA_F32_16X16X4_F32 accums.
// Per K=4 slice: 8 global_load_b64 vs 16 WMMA.
// M, N multiples of 64; Kd multiple of 4.
// ---------------------------------------------------------------------------
__global__ __launch_bounds__(32)
void gemm_wmma_f32(const float* __restrict__ A, const float* __restrict__ Bt,
                   const float* __restrict__ Cin, float* __restrict__ D,
                   int M, int N, int Kd, float alpha, float beta,
                   int tilesN, int tilesPerBatch)
{
    int tile = blockIdx.x;
    int b    = tile / tilesPerBatch;
    int r    = tile - b * tilesPerBatch;
    int tm   = r / tilesN;
    int tn   = r - tm * tilesN;

    long aOff = (long)b * M * Kd;
    long bOff = (long)b * N * Kd;
    long cOff = (long)b * M * N;

    int lane  = threadIdx.x;
    int lrow  = lane & 15;   // 0..15
    int khalf = lane >> 4;   // 0 or 1
    int m0 = tm * 64, n0 = tn * 64;

    v8f acc[4][4] = {};

    // Per-lane fragment base pointers; A lane holds A[m][k0+2*khalf .. +1],
    // Bt lane holds B[k0+2*khalf .. +1][n] == Bt[n][k0+2*khalf .. +1].
    const v2f* Ap[4];
    const v2f* Bp[4];
    #pragma unroll
    for (int i = 0; i < 4; ++i)
        Ap[i] = (const v2f*)(A + aOff + (long)(m0 + i * 16 + lrow) * Kd + 2 * khalf);
    #pragma unroll
    for (int j = 0; j < 4; ++j)
        Bp[j] = (const v2f*)(Bt + bOff + (long)(n0 + j * 16 + lrow) * Kd + 2 * khalf);

    int kv = 0;                       // v2f index; advances 2 per K=4 slice
    #pragma unroll 2
    for (int k0 = 0; k0 < Kd; k0 += 4, kv += 2) {
        v2f af[4], bf[4];
        #pragma unroll
        for (int i = 0; i < 4; ++i) af[i] = Ap[i][kv];
        #pragma unroll
        for (int j = 0; j < 4; ++j) bf[j] = Bp[j][kv];
        #pragma unroll
        for (int i = 0; i < 4; ++i)
            #pragma unroll
            for (int j = 0; j < 4; ++j)
                acc[i][j] = __builtin_amdgcn_wmma_f32_16x16x4_f32(
                    false, af[i], false, bf[j], (short)0, acc[i][j], false, false);
    }

    // C/D layout: VGPR rr -> rows M=rr (lanes 0-15) and M=rr+8 (lanes 16-31), N=lrow
    #pragma unroll
    for (int i = 0; i < 4; ++i)
        #pragma unroll
        for (int j = 0; j < 4; ++j)
            #pragma unroll
            for (int rr = 0; rr < 8; ++rr) {
                int row = m0 + i * 16 + khalf * 8 + rr;
                int col = n0 + j * 16 + lrow;
                long off = cOff + (long)row * N + col;
                float v = alpha * acc[i][j][rr];
                if (beta != 0.0f) v += beta * Cin[off];
                D[off] = v;
            }
}

// ---------------------------------------------------------------------------
// Batched transpose: dst[b][c][r] = src[b][r][c];  src (batch,R,Cc) -> dst (batch,Cc,R)
// ---------------------------------------------------------------------------
__global__ void transpose_k(const float* __restrict__ src, float* __restrict__ dst,
                            int R, int Cc, long n)
{
    long i = (long)blockIdx.x * blockDim.x + threadIdx.x;
    if (i >= n) return;
    long per = (long)R * Cc;
    long b   = i / per;
    long rm  = i - b * per;
    int cc = (int)(rm / R);
    int rr = (int)(rm - (long)cc * R);
    dst[i] = src[b * per + (long)rr * Cc + cc];
}

// ---------------------------------------------------------------------------
// Elementwise / init kernels
// ---------------------------------------------------------------------------
__global__ void fill_zero_k(float* p, int n) {
    int i = blockIdx.x * blockDim.x + threadIdx.x;
    if (i < n) p[i] = 0.0f;
}

__global__ void init_mean_k(float* mean, const float* mm, int n) {
    int i = blockIdx.x * blockDim.x + threadIdx.x;
    if (i < n) mean[i] = mm[i % (KDIM * CDIM)];
}

__global__ void init_cov_k(float* cov, int n) {
    int i = blockIdx.x * blockDim.x + threadIdx.x;
    if (i >= n) return;
    int r = i % (KDIM * KDIM);
    int ii = r / KDIM, jj = r % KDIM;
    cov[i] = (ii == jj) ? (1.0f + 1e-6f) : 0.0f;
}

// Pa = ALPHA * meanT  (meanT already is mean^T, contiguous)
__global__ void pinv_scale_k(const float* __restrict__ meanT, float* __restrict__ P, long n) {
    long i = (long)blockIdx.x * blockDim.x + threadIdx.x;
    if (i < n) P[i] = ALPHA_PINV * meanT[i];
}

// ---------------------------------------------------------------------------
// LSTM kernels (stream-ordered recurrence; both directions in one launch)
// ---------------------------------------------------------------------------
__global__ void lstm_gates_k(const float* __restrict__ xin, int tF, int tB,
                             const float* __restrict__ Wih, const float* __restrict__ Whh,
                             const float* __restrict__ bvec,
                             const float* __restrict__ hprev, float* __restrict__ gbuf)
{
    int idx = blockIdx.x * blockDim.x + threadIdx.x;
    if (idx >= 2 * BDIM * G4) return;
    int d   = idx / (BDIM * G4);
    int rem = idx % (BDIM * G4);
    int b   = rem / G4;
    int gi  = rem % G4;
    int t   = (d == 0) ? tF : tB;

    const float* x  = xin + ((long)t * BDIM + b) * CDIM;
    const float* wi = Wih + ((long)d * G4 + gi) * CDIM;
    float s = bvec[d * G4 + gi];
    for (int c = 0; c < CDIM; ++c) s += x[c] * wi[c];
    const float* h  = hprev + ((long)d * BDIM + b) * HDIM;
    const float* wh = Whh + ((long)d * G4 + gi) * HDIM;
    for (int c = 0; c < HDIM; ++c) s += h[c] * wh[c];
    gbuf[idx] = s;
}

__global__ void lstm_cell_k(const float* __restrict__ gbuf,
                            float* __restrict__ hstate, float* __restrict__ cstate,
                            float* __restrict__ out, int tF, int tB)
{
    int idx = blockIdx.x * blockDim.x + threadIdx.x;
    if (idx >= 2 * BDIM * HDIM) return;
    int d   = idx / (BDIM * HDIM);
    int rem = idx % (BDIM * HDIM);
    int b   = rem / HDIM;
    int hh  = rem % HDIM;

    const float* g = gbuf + ((long)d * BDIM + b) * G4;
    float gi = sigm(g[hh]);
    float gf = sigm(g[HDIM + hh]);
    float gg = tanhf(g[2 * HDIM + hh]);
    float go = sigm(g[3 * HDIM + hh]);

    long sidx = ((long)d * BDIM + b) * HDIM + hh;
    float c = gf * cstate[sidx] + gi * gg;
    float h = go * tanhf(c);
    cstate[sidx] = c;
    hstate[sidx] = h;

    int t = (d == 0) ? tF : tB;
    out[((long)t * BDIM + b) * CDIM + d * HDIM + hh] = h;
}

// ---------------------------------------------------------------------------
// Sherman-Morrison step vector kernels
// ---------------------------------------------------------------------------
// w[b,k] = sum_c (zl[t,b,c] + OBS_STD*noise[t,b,c]) * P[b,c,k]
__global__ void w_kernel(const float* __restrict__ zl, const float* __restrict__ noise,
                         const float* __restrict__ P, float* __restrict__ w, int t)
{
    int idx = blockIdx.x * blockDim.x + threadIdx.x;
    if (idx >= BDIM * KDIM) return;
    int b = idx / KDIM, k = idx % KDIM;
    const float* zp = zl + ((long)t * BDIM + b) * CDIM;
    const float* np = noise + ((long)t * BDIM + b) * CDIM;
    const float* Pp = P + (long)b * CDIM * KDIM + k;
    float s = 0.0f;
    for (int c = 0; c < CDIM; ++c) s += (zp[c] + OBS_STD * np[c]) * Pp[(long)c * KDIM];
    w[idx] = s;
}

// Delta[b,c] = zl[t,b,c] - sum_k w[b,k]*mean[b,k,c]
__global__ void delta_kernel(const float* __restrict__ zl, const float* __restrict__ w,
                             const float* __restrict__ mean, float* __restrict__ Delta, int t)
{
    int idx = blockIdx.x * blockDim.x + threadIdx.x;
    if (idx >= BDIM * CDIM) return;
    int b = idx / CDIM, c = idx % CDIM;
    const float* wp = w + (long)b * KDIM;
    const float* mp = mean + (long)b * KDIM * CDIM + c;
    float s = 0.0f;
    for (int k = 0; k < KDIM; ++k) s += wp[k] * mp[(long)k * CDIM];
    Delta[idx] = zl[((long)t * BDIM + b) * CDIM + c] - s;
}

// wU[b,j] = sum_k w[b,k]*cov[b,k,j]
__global__ void wu_kernel(const float* __restrict__ w, const float* __restrict__ cov,
                          float* __restrict__ wU)
{
    int idx = blockIdx.x * blockDim.x + threadIdx.x;
    if (idx >= BDIM * KDIM) return;
    int b = idx / KDIM, j = idx % KDIM;
    const float* wp = w + (long)b * KDIM;
    const float* cp = cov + (long)b * KDIM * KDIM + j;
    float s = 0.0f;
    for (int k = 0; k < KDIM; ++k) s += wp[k] * cp[(long)k * KDIM];
    wU[idx] = s;
}

__global__ void sigma_kernel(const float* __restrict__ wU, const float* __restrict__ w,
                             float* __restrict__ sigma)
{
    __shared__ float sm[256];
    int b = blockIdx.x;
    float s = 0.0f;
    for (int k = threadIdx.x; k < KDIM; k += 256)
        s += wU[(long)b * KDIM + k] * w[(long)b * KDIM + k];
    sm[threadIdx.x] = s;
    __syncthreads();
    for (int st = 128; st > 0; st >>= 1) {
        if (threadIdx.x < st) sm[threadIdx.x] += sm[threadIdx.x + st];
        __syncthreads();
    }
    if (threadIdx.x == 0)
        sigma[b] = fmaxf(sm[0] + OBS_STD * OBS_STD, 1e-6f);
}

__global__ void cvec_kernel(const float* __restrict__ wU, const float* __restrict__ sigma,
                            float* __restrict__ cvec)
{
    int idx = blockIdx.x * blockDim.x + threadIdx.x;
    if (idx >= BDIM * KDIM) return;
    int b = idx / KDIM;
    cvec[idx] = clampf(wU[idx] / sigma[b], -1000.0f, 1000.0f);
}

// mean = clip(mean + c[b,k]*clip(Delta[b,c], +-100), +-1000)  (in place)
__global__ void mean_update_k(float* __restrict__ mean, const float* __restrict__ cvec,
                              const float* __restrict__ Delta, int n)
{
    int idx = blockIdx.x * blockDim.x + threadIdx.x;
    if (idx >= n) return;
    int b = idx / (KDIM * CDIM);
    int r = idx % (KDIM * CDIM);
    int k = r / CDIM, c = r % CDIM;
    float dc = clampf(Delta[(long)b * CDIM + c], -100.0f, 100.0f);
    float m  = mean[idx] + cvec[(long)b * KDIM + k] * dc;
    mean[idx] = clampf(m, -1000.0f, 1000.0f);
}

// covDst = sym(covSrc - c wU^T), diag clip + EPS
__global__ void cov_update_k(const float* __restrict__ covSrc, const float* __restrict__ cvec,
                             const float* __restrict__ wU, float* __restrict__ covDst, int n)
{
    int idx = blockIdx.x * blockDim.x + threadIdx.x;
    if (idx >= n) return;
    int b = idx / (KDIM * KDIM);
    int r = idx % (KDIM * KDIM);
    int i = r / KDIM, j = r % KDIM;
    long base = (long)b * KDIM * KDIM;
    const float* cv = cvec + (long)b * KDIM;
    const float* wu = wU + (long)b * KDIM;
    float uij = covSrc[base + (long)i * KDIM + j] - cv[i] * wu[j];
    float uji = covSrc[base + (long)j * KDIM + i] - cv[j] * wu[i];
    float v = 0.5f * (uij + uji);
    if (i == j) v = clampf(v, 0.001f, 1000.0f) + 1e-6f;
    covDst[idx] = v;
}

// ---------------------------------------------------------------------------
// KL reduction
// ---------------------------------------------------------------------------
__global__ void kl_t14_k(const float* __restrict__ cov, float* __restrict__ t14)
{
    __shared__ float sr[256], sl[256];
    int b = blockIdx.x;
    const float p = 1.0f + 1e-6f;
    float ar = 0.0f, al = 0.0f;
    for (int k = threadIdx.x; k < KDIM; k += 256) {
        float q = clampf(cov[(long)b * KDIM * KDIM + (long)k * KDIM + k], 0.001f, 1e6f);
        ar += clampf(q / p, 1e-6f, 1000.0f);
        al += clampf(logf(p) - logf(q), -10.0f, 10.0f);
    }
    sr[threadIdx.x] = ar; sl[threadIdx.x] = al;
    __syncthreads();
    for (int st = 128; st > 0; st >>= 1) {
        if (threadIdx.x < st) { sr[threadIdx.x] += sr[threadIdx.x + st];
                                sl[threadIdx.x] += sl[threadIdx.x + st]; }
        __syncthreads();
    }
    if (threadIdx.x == 0)
        t14[b] = clampf((float)CDIM * sr[0], -1e6f, 1e6f) +
                 clampf((float)CDIM * sl[0], -1e6f, 1e6f);
}

__global__ void kl_t2_k(const float* __restrict__ mean, const float* __restrict__ mm,
                        float* __restrict__ t2b)
{
    __shared__ float sm[256];
    int b = blockIdx.x;
    const float pinv = 1.0f / (1.0f + 1e-6f);
    float s = 0.0f;
    const int n = KDIM * CDIM;
    for (int i = threadIdx.x; i < n; i += 256) {
        float diff = mean[(long)b * n + i] - mm[i];
        s += fminf(diff * diff, 1000.0f) * pinv;
    }
    sm[threadIdx.x] = s;
    __syncthreads();
    for (int st = 128; st > 0; st >>= 1) {
        if (threadIdx.x < st) sm[threadIdx.x] += sm[threadIdx.x + st];
        __syncthreads();
    }
    if (threadIdx.x == 0) t2b[b] = clampf(sm[0], -1e6f, 1e6f);
}

__global__ void kl_final_k(const float* __restrict__ t14, const float* __restrict__ t2b,
                           float* __restrict__ dkl)
{
    float acc = 0.0f;
    const float t3 = -(float)CDIM * (float)KDIM;
    for (int b = 0; b < BDIM; ++b) acc += t14[b] + t2b[b] + t3;
    *dkl = acc / (float)BDIM;
}

// ---------------------------------------------------------------------------
// Host driver
// ---------------------------------------------------------------------------
static inline dim3 gridFor(long n, int bs) { return dim3((unsigned)((n + bs - 1) / bs)); }

extern "C" void kernel_launch(void* const* d_in, const int* in_sizes, int n_in,
                              void* d_out, int out_size, void* d_ws, size_t ws_size,
                              hipStream_t stream)
{
    const float* z     = (const float*)d_in[0];
    const float* mm    = (const float*)d_in[1];
    const float* noise = (const float*)d_in[2];
    const float* Wih0  = (const float*)d_in[3];
    const float* Whh0  = (const float*)d_in[4];
    const float* b0    = (const float*)d_in[5];
    const float* Wih1  = (const float*)d_in[6];
    const float* Whh1  = (const float*)d_in[7];
    const float* b1    = (const float*)d_in[8];

    float* meanOut = (float*)d_out;
    float* covOut  = meanOut + (size_t)BDIM * KDIM * CDIM;
    float* dklOut  = covOut + (size_t)BDIM * KDIM * KDIM;

    float* ws = (float*)d_ws;
    size_t off = 0;
    auto alloc = [&](size_t n) { float* p = ws + off; off += n; return p; };

    float* zl    = alloc((size_t)EDIM * BDIM * CDIM);
    float* l0out = alloc((size_t)EDIM * BDIM * CDIM);
    float* gbuf  = alloc((size_t)2 * BDIM * G4);
    float* hst   = alloc((size_t)2 * BDIM * HDIM);
    float* cst   = alloc((size_t)2 * BDIM * HDIM);
    float* Pa    = alloc((size_t)BDIM * CDIM * KDIM);
    float* Pb    = alloc((size_t)BDIM * CDIM * KDIM);
    float* Pt    = alloc((size_t)BDIM * KDIM * CDIM);   // Pcur^T (K-major for GEMM2 B)
    float* meanT = alloc((size_t)BDIM * CDIM * KDIM);   // mean^T (K-major for GEMM1 B)
    float* Tm    = alloc((size_t)BDIM * CDIM * CDIM);
    float* covWs = alloc((size_t)BDIM * KDIM * KDIM);
    float* wv    = alloc((size_t)BDIM * KDIM);
    float* Delta = alloc((size_t)BDIM * CDIM);
    float* wU    = alloc((size_t)BDIM * KDIM);
    float* sigma = alloc((size_t)BDIM);
    float* cvec  = alloc((size_t)BDIM * KDIM);
    float* t14   = alloc((size_t)BDIM);
    float* t2b   = alloc((size_t)BDIM);

    // ---- LSTM (2 layers, bidirectional), stream-ordered recurrence ----
    const float* Wihs[2] = {Wih0, Wih1};
    const float* Whhs[2] = {Whh0, Whh1};
    const float* bs[2]   = {b0, b1};
    const float* lin[2]  = {z, l0out};
    float*       lout[2] = {l0out, zl};

    for (int layer = 0; layer < 2; ++layer) {
        fill_zero_k<<<gridFor(2 * BDIM * HDIM, 256), 256, 0, stream>>>(hst, 2 * BDIM * HDIM);
        fill_zero_k<<<gridFor(2 * BDIM * HDIM, 256), 256, 0, stream>>>(cst, 2 * BDIM * HDIM);
        for (int s = 0; s < EDIM; ++s) {
            lstm_gates_k<<<gridFor(2 * BDIM * G4, 256), 256, 0, stream>>>(
                lin[layer], s, EDIM - 1 - s, Wihs[layer], Whhs[layer], bs[layer], hst, gbuf);
            lstm_cell_k<<<gridFor(2 * BDIM * HDIM, 256), 256, 0, stream>>>(
                gbuf, hst, cst, lout[layer], s, EDIM - 1 - s);
        }
    }

    // ---- init posterior ----
    const long nMean = (long)BDIM * KDIM * CDIM;
    const long nCov  = (long)BDIM * KDIM * KDIM;
    const long nP    = (long)BDIM * CDIM * KDIM;
    init_mean_k<<<gridFor(nMean, 256), 256, 0, stream>>>(meanOut, mm, (int)nMean);
    init_cov_k<<<gridFor(nCov, 256), 256, 0, stream>>>(covOut, (int)nCov);

    // ---- episode steps ----
    for (int t = 0; t < EDIM; ++t) {
        float* covSrc = (t % 2 == 0) ? covOut : covWs;
        float* covDst = (t % 2 == 0) ? covWs : covOut;

        // meanT = mean^T (once per step); P0 = ALPHA * meanT
        transpose_k<<<gridFor(nMean, 256), 256, 0, stream>>>(meanOut, meanT, KDIM, CDIM, nMean);
        pinv_scale_k<<<gridFor(nP, 256), 256, 0, stream>>>(meanT, Pa, nP);

        // 3 Newton-Schulz iterations: T = P@mean ; Pnext = 2P - T@P
        float* Pcur = Pa; float* Pnxt = Pb;
        for (int it = 0; it < 3; ++it) {
            // Pt = Pcur^T (K-major B operand for GEMM2)
            transpose_k<<<gridFor(nP, 256), 256, 0, stream>>>(Pcur, Pt, CDIM, KDIM, nP);
            // T (B,896,896) = Pcur (B,896,512) @ mean (B,512,896); Bt = meanT
            {
                int M = CDIM, N = CDIM, Kd = KDIM;
                int tilesM = M / 64, tilesN = N / 64;
                gemm_wmma_f32<<<dim3(BDIM * tilesM * tilesN), 32, 0, stream>>>(
                    Pcur, meanT, Tm, Tm, M, N, Kd, 1.0f, 0.0f, tilesN, tilesM * tilesN);
            }
            // Pnxt (B,896,512) = -1 * T @ Pcur + 2 * Pcur; Bt = Pt
            {
                int M = CDIM, N = KDIM, Kd = CDIM;
                int tilesM = M / 64, tilesN = N / 64;
                gemm_wmma_f32<<<dim3(BDIM * tilesM * tilesN), 32, 0, stream>>>(
                    Tm, Pt, Pcur, Pnxt, M, N, Kd, -1.0f, 2.0f, tilesN, tilesM * tilesN);
            }
            float* tmp = Pcur; Pcur = Pnxt; Pnxt = tmp;
        }
        // After 3 iters (Pa->Pb->Pa->Pb), Pcur holds the pseudo-inverse.

        w_kernel<<<gridFor(BDIM * KDIM, 256), 256, 0, stream>>>(zl, noise, Pcur, wv, t);
        delta_kernel<<<gridFor(BDIM * CDIM, 256), 256, 0, stream>>>(zl, wv, meanOut, Delta, t);
        wu_kernel<<<gridFor(BDIM * KDIM, 256), 256, 0, stream>>>(wv, covSrc, wU);
        sigma_kernel<<<dim3(BDIM), 256, 0, stream>>>(wU, wv, sigma);
        cvec_kernel<<<gridFor(BDIM * KDIM, 256), 256, 0, stream>>>(wU, sigma, cvec);
        mean_update_k<<<gridFor(nMean, 256), 256, 0, stream>>>(meanOut, cvec, Delta, (int)nMean);
        cov_update_k<<<gridFor(nCov, 256), 256, 0, stream>>>(covSrc, cvec, wU, covDst, (int)nCov);
    }
    // EDIM=4 -> final cov is in covOut (d_out).  mean updated in place in d_out.

    // ---- KL ----
    kl_t14_k<<<dim3(BDIM), 256, 0, stream>>>(covOut, t14);
    kl_t2_k<<<dim3(BDIM), 256, 0, stream>>>(meanOut, mm, t2b);
    kl_final_k<<<dim3(1), 1, 0, stream>>>(t14, t2b, dklOut);

    (void)in_sizes; (void)n_in; (void)out_size; (void)ws_size;
}